// lossFunc_41841571397867
// MI455X (gfx1250) — compile-verified
//
#include <hip/hip_runtime.h>
#include <math.h>

typedef __attribute__((ext_vector_type(2))) float v2f;
typedef __attribute__((ext_vector_type(8))) float v8f;

#define BATCH   64
#define WIDTH   512
#define NPB     (512*512)        // elements per batch = 262144
#define CHUNKS  64               // chunks per batch
#define CHUNK   (NPB/CHUNKS)     // 4096 elements per block
#define TPB     256              // 8 wave32s
#define NITER   (CHUNK/(TPB*4))  // 4 float4 loads per thread
#define NBLOCKS (BATCH*CHUNKS)   // 4096 partial records

// ---------------------------------------------------------------------------
// Kernel 1: fused streaming pass. Per block: partial MSE (via WMMA f32
// accumulate against an all-ones B matrix) + per-chunk argmax of pred & label.
// ---------------------------------------------------------------------------
__global__ __launch_bounds__(TPB) void fused_partial_kernel(
    const float* __restrict__ pred, const float* __restrict__ label,
    float* __restrict__ mse_p,
    float* __restrict__ pmax_p, unsigned* __restrict__ pidx_p,
    float* __restrict__ lmax_p, unsigned* __restrict__ lidx_p)
{
    const int c   = blockIdx.x;           // chunk within batch
    const int b   = blockIdx.y;           // batch
    const int tid = threadIdx.x;
    const long base = (long)b * NPB + (long)c * CHUNK;
    const float4* p4 = (const float4*)(pred  + base);
    const float4* l4 = (const float4*)(label + base);

    v8f acc = {};                          // WMMA C/D accumulator (16x16 f32)
    v2f ones; ones[0] = 1.0f; ones[1] = 1.0f;
    float fallback_acc = 0.0f;             // used only if builtin unavailable

    float pmax = -__builtin_inff(); unsigned pidx = 0u;
    float lmax = -__builtin_inff(); unsigned lidx = 0u;

#pragma unroll
    for (int it = 0; it < NITER; ++it) {
        const int vec = it * TPB + tid;    // float4 index within chunk
        const float4 p = p4[vec];
        const float4 l = l4[vec];
        const unsigned e = (unsigned)(c * CHUNK + vec * 4);  // flat idx in batch

        // argmax (strict > keeps lowest index; e increases within thread)
        if (p.x > pmax) { pmax = p.x; pidx = e;     }
        if (p.y > pmax) { pmax = p.y; pidx = e + 1; }
        if (p.z > pmax) { pmax = p.z; pidx = e + 2; }
        if (p.w > pmax) { pmax = p.w; pidx = e + 3; }
        if (l.x > lmax) { lmax = l.x; lidx = e;     }
        if (l.y > lmax) { lmax = l.y; lidx = e + 1; }
        if (l.z > lmax) { lmax = l.z; lidx = e + 2; }
        if (l.w > lmax) { lmax = l.w; lidx = e + 3; }

        const float d0 = p.x - l.x, d1 = p.y - l.y;
        const float d2 = p.z - l.z, d3 = p.w - l.w;
#if __has_builtin(__builtin_amdgcn_wmma_f32_16x16x4_f32)
        v2f a0; a0[0] = d0 * d0; a0[1] = d1 * d1;
        v2f a1; a1[0] = d2 * d2; a1[1] = d3 * d3;
        // D = A x ones + C : row-sums the 64 squared diffs into the
        // accumulator on the matrix pipe (co-executes with the VALU argmax).
        acc = __builtin_amdgcn_wmma_f32_16x16x4_f32(
            false, a0, false, ones, (short)0, acc, false, false);
        acc = __builtin_amdgcn_wmma_f32_16x16x4_f32(
            false, a1, false, ones, (short)0, acc, false, false);
#else
        fallback_acc += d0 * d0 + d1 * d1 + d2 * d2 + d3 * d3;
#endif
    }

    // Collapse WMMA accumulator -> per-wave scalar.
    // D layout: VGPR r, lanes 0-15 hold row M=r (all cols equal since B=ones),
    // lanes 16-31 hold row M=8+r. So comp-sum gives rows 0-7 (lanes 0-15) and
    // rows 8-15 (lanes 16-31); xor-16 shuffle completes the wave total.
    float s;
#if __has_builtin(__builtin_amdgcn_wmma_f32_16x16x4_f32)
    s = acc[0] + acc[1] + acc[2] + acc[3] + acc[4] + acc[5] + acc[6] + acc[7];
#else
    s = fallback_acc * 16.0f * 0.0f + fallback_acc; // plain per-lane partial
    s += __shfl_xor(s, 1, 32); s += __shfl_xor(s, 2, 32);
    s += __shfl_xor(s, 4, 32); s += __shfl_xor(s, 8, 32);
#endif
    s += __shfl_xor(s, 16, 32);

    __shared__ float    s_mse[TPB / 32];
    __shared__ float    s_pv[TPB]; __shared__ unsigned s_pi[TPB];
    __shared__ float    s_lv[TPB]; __shared__ unsigned s_li[TPB];

    const int lane = tid & 31, wave = tid >> 5;
    if (lane == 0) s_mse[wave] = s;
    s_pv[tid] = pmax; s_pi[tid] = pidx;
    s_lv[tid] = lmax; s_li[tid] = lidx;
    __syncthreads();

    for (int st = TPB / 2; st > 0; st >>= 1) {
        if (tid < st) {
            float v = s_pv[tid + st]; unsigned i = s_pi[tid + st];
            if (v > s_pv[tid] || (v == s_pv[tid] && i < s_pi[tid])) { s_pv[tid] = v; s_pi[tid] = i; }
            v = s_lv[tid + st]; i = s_li[tid + st];
            if (v > s_lv[tid] || (v == s_lv[tid] && i < s_li[tid])) { s_lv[tid] = v; s_li[tid] = i; }
        }
        __syncthreads();
    }

    if (tid == 0) {
        float m = 0.0f;
#pragma unroll
        for (int i = 0; i < TPB / 32; ++i) m += s_mse[i];
        const int bid = b * CHUNKS + c;
        mse_p[bid]  = m;
        pmax_p[bid] = s_pv[0]; pidx_p[bid] = s_pi[0];
        lmax_p[bid] = s_lv[0]; lidx_p[bid] = s_li[0];
    }
}

// ---------------------------------------------------------------------------
// Kernel 2: tiny scalar epilogue (sum partials, combine argmaxes, formula).
// ---------------------------------------------------------------------------
__global__ __launch_bounds__(256) void finalize_kernel(
    const float* __restrict__ mse_p,
    const float* __restrict__ pmax_p, const unsigned* __restrict__ pidx_p,
    const float* __restrict__ lmax_p, const unsigned* __restrict__ lidx_p,
    float* __restrict__ out)
{
    __shared__ float red[256];
    __shared__ float s_mse_total;
    const int tid = threadIdx.x;

    // total MSE over 4096 block partials
    float m = 0.0f;
    for (int i = tid; i < NBLOCKS; i += 256) m += mse_p[i];
    red[tid] = m; __syncthreads();
    for (int st = 128; st > 0; st >>= 1) {
        if (tid < st) red[tid] += red[tid + st];
        __syncthreads();
    }
    if (tid == 0) s_mse_total = red[0];
    __syncthreads();

    // per-batch argmax combine + coordinate loss
    float idxpart = 0.0f;
    if (tid < BATCH) {
        float pv = -__builtin_inff(); unsigned pi = 0u;
        float lv = -__builtin_inff(); unsigned li = 0u;
        for (int cix = 0; cix < CHUNKS; ++cix) {
            const int k = tid * CHUNKS + cix;
            float v = pmax_p[k]; unsigned i = pidx_p[k];
            if (v > pv || (v == pv && i < pi)) { pv = v; pi = i; }
            v = lmax_p[k]; i = lidx_p[k];
            if (v > lv || (v == lv && i < li)) { lv = v; li = i; }
        }
        const float dr = (float)(pi / WIDTH) - (float)(li / WIDTH);
        const float dc = (float)(pi % WIDTH) - (float)(li % WIDTH);
        idxpart = dr * dr + dc * dc;
    }
    __syncthreads();
    red[tid] = idxpart; __syncthreads();
    for (int st = 128; st > 0; st >>= 1) {
        if (tid < st) red[tid] += red[tid + st];
        __syncthreads();
    }

    if (tid == 0) {
        const float idx_loss = red[0];
        const float mse_loss = s_mse_total;
        const float alpha    = (idx_loss != 0.0f) ? (mse_loss / idx_loss) : 1.0f;
        out[0] = (mse_loss + 0.25f * alpha * idx_loss) / (float)BATCH;
    }
}

// ---------------------------------------------------------------------------
extern "C" void kernel_launch(void* const* d_in, const int* in_sizes, int n_in,
                              void* d_out, int out_size, void* d_ws, size_t ws_size,
                              hipStream_t stream) {
    const float* pred  = (const float*)d_in[0];
    const float* label = (const float*)d_in[1];   // (B,1,H,W) == (B,H,W) flat

    float* ws = (float*)d_ws;                     // 5*4096 floats = 80 KB
    float*    mse_p  = ws;
    float*    pmax_p = ws + NBLOCKS;
    unsigned* pidx_p = (unsigned*)(ws + 2 * NBLOCKS);
    float*    lmax_p = ws + 3 * NBLOCKS;
    unsigned* lidx_p = (unsigned*)(ws + 4 * NBLOCKS);

    dim3 grid(CHUNKS, BATCH);
    fused_partial_kernel<<<grid, TPB, 0, stream>>>(pred, label, mse_p,
                                                   pmax_p, pidx_p, lmax_p, lidx_p);
    finalize_kernel<<<1, 256, 0, stream>>>(mse_p, pmax_p, pidx_p,
                                           lmax_p, lidx_p, (float*)d_out);
}